// BlockPositioning_7791070675400
// MI455X (gfx1250) — compile-verified
//
#include <hip/hip_runtime.h>
#include <hip/hip_bf16.h>

// out[b*H+h, sk, sq] = ev(sk/4, sq/4, h) + ch[sk%4, sq%4, h]
//   ev(ek,eq,h) = (ek>eq) ? event_blocks[ek-eq,0,h] : event_blocks_future[eq-ek,0,h]
//
// Store-bandwidth bound: 512 MB out / 23.3 TB/s ~= 22 us. Each 16x16 tile is
// computed exactly with two chained V_WMMA_F32_16X16X4_F32 (one-hot B
// selectors), free behind the HBM store wall. Stores use 32-bit voffset +
// scalar base (output < 4 GB) with immediate row offsets, and NT hints so the
// write-once stream doesn't churn L2.

typedef __attribute__((ext_vector_type(2))) float v2f;
typedef __attribute__((ext_vector_type(8))) float v8f;

#define H_   8
#define EK_  512
#define SK_  2048
#define SQ_  2048

template <int NB>
__global__ __launch_bounds__(256) void bp_wmma_kernel(
    const float* __restrict__ channel_blocks,   // (4,4,1,8)
    const float* __restrict__ event_blocks,     // (512,1,8)
    const float* __restrict__ event_future,     // (512,1,8)
    float* __restrict__ out)
{
    __shared__ float s_past[EK_];
    __shared__ float s_fut[EK_];

    const int h   = blockIdx.x >> 7;      // head 0..7
    const int tsk = blockIdx.x & 127;     // sk tile 0..127
    const int sk_base = tsk * 16;
    const int ek_base = tsk * 4;

    // Stage per-head event tables (strided gather -> LDS), 4 KB total.
    for (int e = threadIdx.x; e < EK_; e += 256) {
        s_past[e] = event_blocks[e * H_ + h];
        s_fut[e]  = event_future[e * H_ + h];
    }
    __syncthreads();

    const int lane = threadIdx.x & 31;    // wave32
    const int wave = threadIdx.x >> 5;    // 0..7
    const int m    = lane & 15;
    const int half = lane >> 4;
    const int k0   = half * 2;            // K slots held by this lane (VGPR0/1)
    const int k1   = k0 + 1;

    // Loop-invariant WMMA operands.
    // B1[j,n] = (n/4 == j) selects eq_local; B2[j,n] = (n%4 == j) selects cq.
    v2f b1, b2, a2;
    b1.x = ((m >> 2) == k0) ? 1.0f : 0.0f;
    b1.y = ((m >> 2) == k1) ? 1.0f : 0.0f;
    b2.x = ((m & 3) == k0) ? 1.0f : 0.0f;
    b2.y = ((m & 3) == k1) ? 1.0f : 0.0f;
    // A2[m,j] = ch[m%4, j, h]   (flat (ck*4+cq)*8 + h)
    const int ck = m & 3;
    a2.x = channel_blocks[(ck * 4 + k0) * H_ + h];
    a2.y = channel_blocks[(ck * 4 + k1) * H_ + h];

    // d0 = e0 - eq_base ; d1 = e0 - 1 - eq_base   (hoisted lane constants)
    const int e0 = ek_base + (m >> 2) - k0;

    // 32-bit element offsets from `out` (total output < 2^30 elements).
    const unsigned row_off =
        (unsigned)((h * SK_ + sk_base + half * 8) * SQ_ + m);
    unsigned boff[NB];
#pragma unroll
    for (int b = 0; b < NB; ++b)
        boff[b] = row_off + (unsigned)b * (unsigned)(H_ * SK_ * SQ_);

    // Each wave covers 16 of the 128 sq tiles.
    for (int tq = wave; tq < 128; tq += 8) {
        const int eq_base = tq * 4;
        const int d0 = e0 - eq_base;
        const int d1 = d0 - 1;
        const float* p0 = (d0 > 0) ? (s_past + d0) : (s_fut - d0);
        const float* p1 = (d1 > 0) ? (s_past + d1) : (s_fut - d1);
        v2f a1;
        a1.x = *p0;   // A1[m,k0] = ev(ek, eq_base+k0)
        a1.y = *p1;   // A1[m,k1]

        // D = A1*B1 + A2*B2  (two chained v_wmma_f32_16x16x4_f32, exact)
        v8f acc = {};
        acc = __builtin_amdgcn_wmma_f32_16x16x4_f32(
                  false, a1, false, b1, (short)0, acc, false, false);
        acc = __builtin_amdgcn_wmma_f32_16x16x4_f32(
                  false, a2, false, b2, (short)0, acc, false, false);

        // Write the tile to every batch replica: SADDR-form NT stores with
        // 32-bit voffset; g*SQ_ (8 KB) folds into the immediate offset.
        const unsigned sq_base = (unsigned)(tq * 16);
#pragma unroll
        for (int b = 0; b < NB; ++b) {
            const unsigned o = boff[b] + sq_base;
#pragma unroll
            for (int g = 0; g < 8; ++g) {
                __builtin_nontemporal_store(acc[g], out + (o + (unsigned)(g * SQ_)));
            }
        }
    }
}

extern "C" void kernel_launch(void* const* d_in, const int* in_sizes, int n_in,
                              void* d_out, int out_size, void* d_ws, size_t ws_size,
                              hipStream_t stream) {
    // inputs: q(f32, values unused), channel_blocks, event_blocks, event_blocks_future
    const float* cb = (const float*)d_in[1];
    const float* eb = (const float*)d_in[2];
    const float* ef = (const float*)d_in[3];
    float* out = (float*)d_out;

    // q shape: (batch*8, 2048, 64) -> batch from flat size
    const int batch = in_sizes[0] / (H_ * 2048 * 64);

    dim3 grid(H_ * 128);   // head x sk-tile
    dim3 block(256);       // 8 wave32

    if (batch == 4) {
        bp_wmma_kernel<4><<<grid, block, 0, stream>>>(cb, eb, ef, out);
    } else {
        // Generic fallback: one launch per batch replica.
        for (int b = 0; b < batch; ++b) {
            bp_wmma_kernel<1><<<grid, block, 0, stream>>>(
                cb, eb, ef, out + (size_t)b * H_ * SK_ * SQ_);
        }
    }
}